// GNNModel_53042846106050
// MI455X (gfx1250) — compile-verified
//
#include <hip/hip_runtime.h>
#include <hip/hip_bf16.h>

typedef __attribute__((ext_vector_type(2))) float v2f;
typedef __attribute__((ext_vector_type(8))) float v8f;

#define GCN_N 50000
#define GCN_E 800000

// ---------------- utility kernels ----------------

__global__ void gcn_zero_kernel(float* __restrict__ p, int n) {
    int i = blockIdx.x * blockDim.x + threadIdx.x;
    if (i < n) p[i] = 0.0f;
}

__global__ void gcn_degree_kernel(const int* __restrict__ dst, float* __restrict__ deg, int E) {
    int e = blockIdx.x * blockDim.x + threadIdx.x;
    if (e < E) atomicAdd(&deg[dst[e]], 1.0f);
}

// deg -> dinv in place: dinv = rsqrt(deg + 1)  (self-loop included)
__global__ void gcn_dinv_kernel(float* __restrict__ deg, int n) {
    int i = blockIdx.x * blockDim.x + threadIdx.x;
    if (i < n) deg[i] = rsqrtf(deg[i] + 1.0f);
}

// ---- WMMA GEMM + fused self-loop/bias epilogue ------------------------------
//   hw  = act(A) @ W
//   agg = hw * dinv^2 + bias          (scatter kernel then adds edge messages)
// One wave per 16x16 output tile, V_WMMA_F32_16X16X4_F32 (full fp32).
// M must be a multiple of 16 (50000 = 3125*16). The main K-loop has NO
// conditional loads; the K-tail and narrow-Nc cases use clamped-index loads
// plus value selects so EXEC stays all-ones everywhere around the WMMAs.
template <int RELU>
__global__ void gcn_gemm_wmma_f32(const float* __restrict__ A,
                                  const float* __restrict__ W,
                                  const float* __restrict__ bias,
                                  const float* __restrict__ dinv,
                                  float* __restrict__ hw,
                                  float* __restrict__ agg,
                                  int M, int K, int Nc) {
    const int lane = threadIdx.x & 31;
    const int wid  = threadIdx.x >> 5;
    const int wavesPerBlock = blockDim.x >> 5;

    const int tilesN = (Nc + 15) >> 4;
    const int tilesM = M >> 4;                 // M is a multiple of 16
    int tile = blockIdx.x * wavesPerBlock + wid;
    if (tile >= tilesM * tilesN) return;       // wave-uniform exit

    const int tm   = tile / tilesN;
    const int tn   = tile - tm * tilesN;
    const int half = lane >> 4;                // 0: K=k,k+1   1: K=k+2,k+3
    const int l16  = lane & 15;

    const int  row   = tm * 16 + l16;          // always < M
    const int  col   = tn * 16 + l16;
    const bool colOK = (col < Nc);
    const int  colc  = colOK ? col : 0;        // clamped -> loads always legal
    const int  koff  = half * 2;

    const float* __restrict__ Arow = A + (size_t)row * K + koff;
    const float* __restrict__ Wcol = W + colc;

    v8f acc = {};
    const int Kmain = K & ~3;

#pragma unroll 4
    for (int k = 0; k < Kmain; k += 4) {
        float a0 = Arow[k];
        float a1 = Arow[k + 1];
        if (RELU) { a0 = fmaxf(a0, 0.0f); a1 = fmaxf(a1, 0.0f); }
        float w0 = Wcol[(size_t)(k + koff)     * Nc];
        float w1 = Wcol[(size_t)(k + koff + 1) * Nc];
        v2f a; a[0] = a0;               a[1] = a1;
        v2f b; b[0] = colOK ? w0 : 0.f; b[1] = colOK ? w1 : 0.f;
        acc = __builtin_amdgcn_wmma_f32_16x16x4_f32(
            false, a, false, b, (short)0, acc, false, false);
    }

    if (Kmain < K) {                           // K % 4 != 0 (layer 0: K=165)
        const int  ka  = Kmain + koff;
        const bool ok0 = (ka     < K);
        const bool ok1 = (ka + 1 < K);
        const int  kc0 = ok0 ? ka     : (K - 1);
        const int  kc1 = ok1 ? ka + 1 : (K - 1);
        float a0 = A[(size_t)row * K + kc0];
        float a1 = A[(size_t)row * K + kc1];
        if (RELU) { a0 = fmaxf(a0, 0.0f); a1 = fmaxf(a1, 0.0f); }
        float w0 = Wcol[(size_t)kc0 * Nc];
        float w1 = Wcol[(size_t)kc1 * Nc];
        v2f a; a[0] = ok0 ? a0 : 0.f;            a[1] = ok1 ? a1 : 0.f;
        v2f b; b[0] = (colOK && ok0) ? w0 : 0.f; b[1] = (colOK && ok1) ? w1 : 0.f;
        acc = __builtin_amdgcn_wmma_f32_16x16x4_f32(
            false, a, false, b, (short)0, acc, false, false);
    }

    // Epilogue: D VGPR r -> M = tm*16 + half*8 + r, N = col.
    const int   rbase = tm * 16 + (half << 3);
    const float bn    = bias[colc];
    if (colOK) {
#pragma unroll
        for (int r = 0; r < 8; ++r) {
            const int m  = rbase + r;
            const float di = dinv[m];
            const float v  = acc[r];
            hw [(size_t)m * Nc + col] = v;
            agg[(size_t)m * Nc + col] = v * di * di + bn;
        }
    }
}

// ---------------- edge scatter: agg[dst] += hw[src] * dinv[src]*dinv[dst] ----
// One wave per edge; 32 lanes stride across C channels (coalesced 128B bursts).
// hw and agg (19.2 MB each) are L2-resident on MI455X (192 MB L2), so both the
// gather reads and the float atomics resolve in L2.
__global__ void gcn_scatter_kernel(const int* __restrict__ src,
                                   const int* __restrict__ dst,
                                   const float* __restrict__ hw,
                                   const float* __restrict__ dinv,
                                   float* __restrict__ agg, int E, int C) {
    int wave = (int)((blockIdx.x * (size_t)blockDim.x + threadIdx.x) >> 5);
    int lane = threadIdx.x & 31;
    if (wave >= E) return;
    int s = src[wave];
    int d = dst[wave];
    float norm = dinv[s] * dinv[d];
    const float* __restrict__ hrow = hw  + (size_t)s * C;
    float*       __restrict__ arow = agg + (size_t)d * C;
    for (int c = lane; c < C; c += 32) {
        atomicAdd(&arow[c], hrow[c] * norm);
    }
}

// ---------------- launch ----------------

extern "C" void kernel_launch(void* const* d_in, const int* in_sizes, int n_in,
                              void* d_out, int out_size, void* d_ws, size_t ws_size,
                              hipStream_t stream) {
    (void)in_sizes; (void)n_in; (void)out_size; (void)ws_size;

    const int N = GCN_N;
    const int E = GCN_E;

    const float* x    = (const float*)d_in[0];
    const int*   ei   = (const int*)d_in[1];    // [2, E] flattened
    const int*   src  = ei;
    const int*   dstp = ei + E;

    const float* W[5] = { (const float*)d_in[2], (const float*)d_in[4],
                          (const float*)d_in[6], (const float*)d_in[8],
                          (const float*)d_in[10] };
    const float* B[5] = { (const float*)d_in[3], (const float*)d_in[5],
                          (const float*)d_in[7], (const float*)d_in[9],
                          (const float*)d_in[11] };

    // workspace layout (floats): dinv[N] | hw[N*96] | aggA[N*96] | aggB[N*96]
    float* ws   = (float*)d_ws;
    float* dinv = ws;
    float* hw   = dinv + N;
    float* aggA = hw   + (size_t)N * 96;
    float* aggB = aggA + (size_t)N * 96;
    float* out  = (float*)d_out;

    const int T = 256;

    // degrees -> dinv
    gcn_zero_kernel  <<<(N + T - 1) / T, T, 0, stream>>>(dinv, N);
    gcn_degree_kernel<<<(E + T - 1) / T, T, 0, stream>>>(dstp, dinv, E);
    gcn_dinv_kernel  <<<(N + T - 1) / T, T, 0, stream>>>(dinv, N);

    const int dims[6] = { 165, 96, 96, 96, 96, 2 };
    const float* layerIn = x;
    float* aggBufs[2] = { aggA, aggB };

    for (int l = 0; l < 5; ++l) {
        const int K  = dims[l];
        const int Cn = dims[l + 1];
        float* agg = (l == 4) ? out : aggBufs[l & 1];

        // GEMM + fused (self-loop + bias) epilogue
        const int tiles         = (N / 16) * ((Cn + 15) / 16);
        const int wavesPerBlock = 4;               // 128 threads
        const int gBlocks       = (tiles + wavesPerBlock - 1) / wavesPerBlock;
        if (l == 0) {
            gcn_gemm_wmma_f32<0><<<gBlocks, wavesPerBlock * 32, 0, stream>>>(
                layerIn, W[l], B[l], dinv, hw, agg, N, K, Cn);
        } else {
            gcn_gemm_wmma_f32<1><<<gBlocks, wavesPerBlock * 32, 0, stream>>>(
                layerIn, W[l], B[l], dinv, hw, agg, N, K, Cn);
        }

        // agg[dst] += hw[src] * dinv[src] * dinv[dst]   (one wave per edge)
        const size_t scatterThreads = (size_t)E * 32;
        gcn_scatter_kernel<<<(scatterThreads + T - 1) / T, T, 0, stream>>>(
            src, dstp, hw, dinv, agg, E, Cn);

        layerIn = agg;
    }
}